// ARTLayer_67843303407661
// MI455X (gfx1250) — compile-verified
//
#include <hip/hip_runtime.h>
#include <math.h>

// Problem constants (K=512, D=512, H=512, PD=128, T=K-1=511)
#define KN   512
#define DN   512
#define HN   512
#define PDN  128
#define TN   511
#define KC   644   // concat K for messages GEMM: 512 (alpha) + 128 (apf) + 1 (asum) + 3 pad

typedef float v2f __attribute__((ext_vector_type(2)));
typedef float v8f __attribute__((ext_vector_type(8)));

// ---------------------------------------------------------------------------
// Generic fp32 WMMA GEMM: C[m,n] = scale * sum_k A[m,k]*Blog[k,n] (+bias[n]) (+addend[m,n]) (relu?)
// BT=true  : Blog[k,n] = B[n*ldb + k]   (i.e. C = A @ B^T, weights stored row-major [N,K])
// BT=false : Blog[k,n] = B[k*ldb + n]
// One wave (32 threads) computes one 16x16 tile, stepping K by 4 with
// V_WMMA_F32_16X16X4_F32.
//   A frag (16x4 f32): lanes 0-15 -> M=lane, {v0,v1}={K+0,K+1};
//                      lanes 16-31 -> M=lane-16, {v0,v1}={K+2,K+3}
//   B frag (4x16 f32): mirrored, lane -> N, halves select K pairs.
//   C/D (16x16 f32, 8 VGPRs): VGPR i holds M=i (lanes 0-15) / M=i+8 (lanes 16-31), N=lane&15.
// ---------------------------------------------------------------------------
template <bool BT, bool RELU>
__global__ __launch_bounds__(32) void gemm_wmma_f32(
    const float* __restrict__ A, int lda,
    const float* __restrict__ B, int ldb,
    const float* __restrict__ bias,     // per-column bias or nullptr
    const float* __restrict__ addend,   // per-element addend (ld = ldc) or nullptr
    float scale,
    float* __restrict__ C, int ldc, int Kd)
{
    const int lane = threadIdx.x;
    const int half = lane >> 4;       // 0: K+{0,1}, 1: K+{2,3}
    const int r    = lane & 15;
    const int m0   = blockIdx.y << 4;
    const int n0   = blockIdx.x << 4;

    const float* arow = A + (size_t)(m0 + r) * lda;
    const float* brow = BT ? (B + (size_t)(n0 + r) * ldb) : nullptr;

    v8f acc = {0.f, 0.f, 0.f, 0.f, 0.f, 0.f, 0.f, 0.f};

    for (int k0 = 0; k0 < Kd; k0 += 4) {
        const int ka = k0 + 2 * half;
        v2f a, b;
        a.x = arow[ka];
        a.y = arow[ka + 1];
        if (BT) {
            b.x = brow[ka];
            b.y = brow[ka + 1];
        } else {
            b.x = B[(size_t)ka * ldb + n0 + r];
            b.y = B[(size_t)(ka + 1) * ldb + n0 + r];
        }
        acc = __builtin_amdgcn_wmma_f32_16x16x4_f32(
            /*neg_a=*/false, a, /*neg_b=*/false, b,
            /*c_mod=*/(short)0, acc, /*reuse_a=*/false, /*reuse_b=*/false);
    }

    const int col = n0 + r;
    const float bv = bias ? bias[col] : 0.f;
#pragma unroll
    for (int i = 0; i < 8; ++i) {
        const int row = m0 + i + 8 * half;
        float v = acc[i] * scale + bv;
        if (addend) v += addend[(size_t)row * ldc + col];
        if (RELU) v = v > 0.f ? v : 0.f;
        C[(size_t)row * ldc + col] = v;
    }
}

// ---------------------------------------------------------------------------
// Zero Acat (incl. diagonal + pad cols), fill Bcat rows 512..643:
//   rows 512..639 = W_pair^T, row 640 = b_pair, rows 641..643 = 0.
// (Bcat rows 0..511 = oj are written by the oj GEMM.)
// ---------------------------------------------------------------------------
__global__ void init_kernel(float* __restrict__ Acat, float* __restrict__ Bcat,
                            const float* __restrict__ W_pair,
                            const float* __restrict__ b_pair)
{
    const int stride = gridDim.x * blockDim.x;
    const int idx = blockIdx.x * blockDim.x + threadIdx.x;
    for (int i = idx; i < KN * KC; i += stride) Acat[i] = 0.f;
    for (int i = idx; i < (KC - KN) * HN; i += stride) {
        const int rr = (i >> 9) + KN;   // 512..643
        const int h  = i & (HN - 1);
        float v = 0.f;
        if (rr < KN + PDN)      v = W_pair[h * PDN + (rr - KN)];
        else if (rr == KN + PDN) v = b_pair[h];
        Bcat[(size_t)rr * HN + h] = v;
    }
}

// ---------------------------------------------------------------------------
// sa[k] = x[k,:] . W_att[0, 0:512];  sb[k] = x[k,:] . W_att[0, 512:1024]
// one wave per row
// ---------------------------------------------------------------------------
__global__ __launch_bounds__(256) void sasb_kernel(
    const float* __restrict__ x, const float* __restrict__ W_att,
    float* __restrict__ sa, float* __restrict__ sb)
{
    const int lane = threadIdx.x & 31;
    const int wave = threadIdx.x >> 5;
    const int row  = blockIdx.x * 8 + wave;
    const float* xr = x + (size_t)row * DN;
    float a = 0.f, b = 0.f;
    for (int i = lane; i < DN; i += 32) {
        const float xv = xr[i];
        a += xv * W_att[i];
        b += xv * W_att[DN + i];
    }
    for (int off = 16; off; off >>= 1) {
        a += __shfl_xor(a, off, 32);
        b += __shfl_xor(b, off, 32);
    }
    if (lane == 0) { sa[row] = a; sb[row] = b; }
}

// ---------------------------------------------------------------------------
// Fused streaming pass over pair_feats (the 134 MB bandwidth-bound stage).
// One block per k (8 wave32s). Each wave owns one t-row of 128 floats per
// iteration (float4 per lane, fully coalesced b128 loads):
//   sc    = pf_row . wc            (butterfly shuffle reduction)
//   alpha = sigmoid(sa[k] + sb[j] + sc + b_att),  j = t + (t >= k)
//   Acat[k, j]        = alpha          (alpha matrix, diag stays 0)
//   Acat[k, 512+p]   += alpha * pf_row[p]   (apf, via LDS ds_add_f32)
//   Acat[k, 640]      = sum_t alpha         (asum)
// ---------------------------------------------------------------------------
__global__ __launch_bounds__(256) void pair_pass(
    const float* __restrict__ pf, const float* __restrict__ W_att,
    const float* __restrict__ sa, const float* __restrict__ sb,
    const float* __restrict__ b_att, float* __restrict__ Acat)
{
    const int k    = blockIdx.x;
    const int lane = threadIdx.x & 31;
    const int wave = threadIdx.x >> 5;

    __shared__ float s_apf[PDN];
    __shared__ float s_asum;
    if (threadIdx.x < PDN) s_apf[threadIdx.x] = 0.f;
    if (threadIdx.x == 0) s_asum = 0.f;
    __syncthreads();

    const float4 wc = *(const float4*)(W_att + 2 * DN + lane * 4);
    const float sak = sa[k] + b_att[0];

    float* __restrict__ arow = Acat + (size_t)k * KC;
    const float* __restrict__ pk = pf + (size_t)k * TN * PDN;

    float4 apf = make_float4(0.f, 0.f, 0.f, 0.f);
    float  asum = 0.f;

    for (int t = wave; t < TN; t += 8) {
        const float4 p = *(const float4*)(pk + (size_t)t * PDN + lane * 4);
        if (t + 8 < TN)
            __builtin_prefetch(pk + (size_t)(t + 8) * PDN + lane * 4, 0, 1);

        float d = p.x * wc.x + p.y * wc.y + p.z * wc.z + p.w * wc.w;
        for (int off = 16; off; off >>= 1) d += __shfl_xor(d, off, 32);

        const int j = t + (t >= k);
        const float alpha = 1.f / (1.f + expf(-(sak + sb[j] + d)));

        if (lane == 0) { arow[j] = alpha; asum += alpha; }
        apf.x += alpha * p.x;
        apf.y += alpha * p.y;
        apf.z += alpha * p.z;
        apf.w += alpha * p.w;
    }

    atomicAdd(&s_apf[lane * 4 + 0], apf.x);
    atomicAdd(&s_apf[lane * 4 + 1], apf.y);
    atomicAdd(&s_apf[lane * 4 + 2], apf.z);
    atomicAdd(&s_apf[lane * 4 + 3], apf.w);
    if (lane == 0) atomicAdd(&s_asum, asum);
    __syncthreads();

    if (threadIdx.x < PDN) arow[KN + threadIdx.x] = s_apf[threadIdx.x];
    if (threadIdx.x == 0)  arow[KN + PDN] = s_asum;
}

// ---------------------------------------------------------------------------
// Row LayerNorm (512 cols, 256 threads/row, 2 elems/thread)
// ---------------------------------------------------------------------------
__global__ __launch_bounds__(256) void ln_kernel(
    const float* __restrict__ Y, const float* __restrict__ g,
    const float* __restrict__ b, float* __restrict__ out)
{
    const int row = blockIdx.x;
    const int tid = threadIdx.x;
    const int lane = tid & 31, wave = tid >> 5;
    const float v0 = Y[(size_t)row * HN + tid];
    const float v1 = Y[(size_t)row * HN + tid + 256];

    __shared__ float red[8];
    float s = v0 + v1;
    for (int off = 16; off; off >>= 1) s += __shfl_xor(s, off, 32);
    if (lane == 0) red[wave] = s;
    __syncthreads();
    float tot = 0.f;
#pragma unroll
    for (int i = 0; i < 8; ++i) tot += red[i];
    const float mean = tot * (1.f / HN);
    const float d0 = v0 - mean, d1 = v1 - mean;
    __syncthreads();

    s = d0 * d0 + d1 * d1;
    for (int off = 16; off; off >>= 1) s += __shfl_xor(s, off, 32);
    if (lane == 0) red[wave] = s;
    __syncthreads();
    tot = 0.f;
#pragma unroll
    for (int i = 0; i < 8; ++i) tot += red[i];
    const float inv = rsqrtf(tot * (1.f / HN) + 1e-5f);

    out[(size_t)row * HN + tid]       = g[tid]       * d0 * inv + b[tid];
    out[(size_t)row * HN + tid + 256] = g[tid + 256] * d1 * inv + b[tid + 256];
}

// ---------------------------------------------------------------------------
extern "C" void kernel_launch(void* const* d_in, const int* in_sizes, int n_in,
                              void* d_out, int out_size, void* d_ws, size_t ws_size,
                              hipStream_t stream)
{
    (void)in_sizes; (void)n_in; (void)out_size; (void)ws_size;
    const float* x      = (const float*)d_in[0];
    const float* pf     = (const float*)d_in[1];
    const float* W_att  = (const float*)d_in[2];
    const float* b_att  = (const float*)d_in[3];
    const float* W_obj  = (const float*)d_in[4];
    const float* b_obj  = (const float*)d_in[5];
    const float* W_pair = (const float*)d_in[6];
    const float* b_pair = (const float*)d_in[7];
    const float* ln_g   = (const float*)d_in[8];
    const float* ln_b   = (const float*)d_in[9];
    const float* W1     = (const float*)d_in[10];
    const float* b1     = (const float*)d_in[11];
    const float* W2     = (const float*)d_in[12];
    const float* b2     = (const float*)d_in[13];

    // workspace layout (floats); total ~5.8 MB
    float* ws   = (float*)d_ws;
    float* sa   = ws;                       // 512
    float* sb   = ws + 512;                 // 512
    float* Acat = ws + 1024;                // 512 x 644
    float* Bcat = Acat + KN * KC;           // 644 x 512
    float* y    = Bcat + KC * HN;           // 512 x 512 (pre-LN residual, reused)
    float* out1 = y    + KN * HN;           // 512 x 512
    float* h1   = out1 + KN * HN;           // 512 x 512

    const dim3 g16(HN / 16, KN / 16);       // (32, 32) tiles

    // 1) zero Acat; fill Bcat static rows (W_pair^T, b_pair, pad)
    init_kernel<<<256, 256, 0, stream>>>(Acat, Bcat, W_pair, b_pair);

    // 2) sa, sb attention projections
    sasb_kernel<<<KN / 8, 256, 0, stream>>>(x, W_att, sa, sb);

    // 3) oj = x @ W_obj^T + b_obj  -> Bcat rows 0..511
    gemm_wmma_f32<true, false><<<g16, 32, 0, stream>>>(
        x, DN, W_obj, DN, b_obj, nullptr, 1.f, Bcat, HN, DN);

    // 4) fused streaming pass over pair_feats -> alpha matrix, apf, asum
    pair_pass<<<KN, 256, 0, stream>>>(pf, W_att, sa, sb, b_att, Acat);

    // 5) y = x + (Acat @ Bcat) / 511   (messages + residual in one GEMM)
    gemm_wmma_f32<false, false><<<g16, 32, 0, stream>>>(
        Acat, KC, Bcat, HN, nullptr, x, 1.f / 511.f, y, HN, KC);

    // 6) out1 = LN(y)
    ln_kernel<<<KN, 256, 0, stream>>>(y, ln_g, ln_b, out1);

    // 7) h1 = relu(out1 @ W1^T + b1)
    gemm_wmma_f32<true, true><<<g16, 32, 0, stream>>>(
        out1, HN, W1, HN, b1, nullptr, 1.f, h1, HN, HN);

    // 8) y = out1 + h1 @ W2^T + b2
    gemm_wmma_f32<true, false><<<g16, 32, 0, stream>>>(
        h1, HN, W2, HN, b2, out1, 1.f, y, HN, HN);

    // 9) out = LN(y)
    ln_kernel<<<KN, 256, 0, stream>>>(y, ln_g, ln_b, (float*)d_out);
}